// Class_Align_Loss_28243704938545
// MI455X (gfx1250) — compile-verified
//
#include <hip/hip_runtime.h>
#include <math.h>

typedef __attribute__((ext_vector_type(2))) float v2f;
typedef __attribute__((ext_vector_type(8))) float v8f;

#define NCLASS 19
#define CHN    256
#define HW     16384          // h*w = 128*128
#define NB     4
#define NPIX   (NB * HW)      // 65536
#define EPS_N  1e-12f
#define EPS_T  1e-6f
#define MARGIN 0.2f
#define LSTRIDE 257           // 19 class rows -> distinct LDS banks per channel
#define NROWS  (2 * NCLASS)   // source rows 0..18, convert rows 19..37

// ws layout (floats)
#define ACC_OFF 0                        // [38][256] raw class sums
#define CEN_OFF (NROWS * CHN)            // [38][256] normalized centers
#define SS_OFF  (CEN_OFF + NROWS * CHN)  // [38] sum of squares of centers
#define S1_OFF  (SS_OFF + NROWS)         // [38] sums of centers

__global__ void __launch_bounds__(256) zero_acc(float* __restrict__ ws) {
  int i = blockIdx.x * blockDim.x + threadIdx.x;
  if (i < NROWS * CHN) ws[ACC_OFF + i] = 0.0f;
}

// One thread per pixel. Pass 1: sum of squares (HBM stream, warms L2).
// Pass 2: re-read (L2-resident: 128 MiB < 192 MB), scale by inv-norm, and
// scatter-add into per-WG LDS class accumulators with ds_add_f32.
__global__ void __launch_bounds__(256) norm_scatter(const float* __restrict__ fs,
                                                    const float* __restrict__ fc,
                                                    const int*   __restrict__ seg,
                                                    float*       __restrict__ ws) {
  __shared__ float lsum[NROWS * LSTRIDE];  // ~39 KB
  for (int i = threadIdx.x; i < NROWS * LSTRIDE; i += blockDim.x) lsum[i] = 0.0f;
  __syncthreads();

  int pid = blockIdx.x * blockDim.x + threadIdx.x;       // exactly 65536 threads
  if (pid < NPIX) {
    int b = pid >> 14;
    int p = pid & (HW - 1);
    const float* ps = fs + (size_t)b * CHN * HW + p;     // lane-contiguous in p
    const float* pc = fc + (size_t)b * CHN * HW + p;

    float ss = 0.0f, tt = 0.0f;
#pragma unroll 8
    for (int c = 0; c < CHN; ++c) {
      float v = ps[(size_t)c * HW];
      float u = pc[(size_t)c * HW];
      ss = fmaf(v, v, ss);
      tt = fmaf(u, u, tt);
    }
    float is = 1.0f / fmaxf(sqrtf(ss), EPS_N);
    float ic = 1.0f / fmaxf(sqrtf(tt), EPS_N);

    int k = seg[pid];
    float* rs = lsum + k * LSTRIDE;
    float* rc = lsum + (NCLASS + k) * LSTRIDE;
#pragma unroll 4
    for (int c = 0; c < CHN; ++c) {
      atomicAdd(&rs[c], ps[(size_t)c * HW] * is);        // ds_add_f32
      atomicAdd(&rc[c], pc[(size_t)c * HW] * ic);
    }
  }
  __syncthreads();

  float* acc = ws + ACC_OFF;
  for (int i = threadIdx.x; i < NROWS * CHN; i += blockDim.x) {
    int r = i >> 8, c = i & 255;
    atomicAdd(&acc[i], lsum[r * LSTRIDE + c]);           // global atomic flush
  }
}

// Divide by n, L2-normalize each 256-wide center row, record sum-of-squares
// and row sums (needed for the eps-expanded distance). Wave-per-row (wave32).
__global__ void __launch_bounds__(256) centers(float* __restrict__ ws) {
  const float invn = 1.0f / (float)NPIX;
  int wave = threadIdx.x >> 5;
  int lane = threadIdx.x & 31;
  for (int r = wave; r < NROWS; r += 8) {
    const float* arow = ws + ACC_OFF + r * CHN;
    float m[8];
    float n2 = 0.0f;
#pragma unroll
    for (int i = 0; i < 8; ++i) {
      m[i] = arow[lane + 32 * i] * invn;
      n2 = fmaf(m[i], m[i], n2);
    }
#pragma unroll
    for (int off = 16; off > 0; off >>= 1) n2 += __shfl_xor(n2, off, 32);
    float inv = 1.0f / fmaxf(sqrtf(n2), EPS_N);

    float* crow = ws + CEN_OFF + r * CHN;
    float s2 = 0.0f, s1 = 0.0f;
#pragma unroll
    for (int i = 0; i < 8; ++i) {
      float o = m[i] * inv;
      crow[lane + 32 * i] = o;
      s2 = fmaf(o, o, s2);
      s1 += o;
    }
#pragma unroll
    for (int off = 16; off > 0; off >>= 1) {
      s2 += __shfl_xor(s2, off, 32);
      s1 += __shfl_xor(s1, off, 32);
    }
    if (lane == 0) {
      ws[SS_OFF + r] = s2;
      ws[S1_OFF + r] = s1;
    }
  }
}

// Gram matrix of the (zero-padded) 32x256 center sets via V_WMMA_F32_16X16X4_F32,
// then the 342-triplet margin loss. 4 waves = 2x2 tiles of 16x16.
__global__ void __launch_bounds__(128) gram_loss(const float* __restrict__ ws,
                                                 float* __restrict__ out) {
  __shared__ float G[32 * 33];
  __shared__ float part[128];

  int wave = threadIdx.x >> 5;
  int lane = threadIdx.x & 31;
  int ti = wave >> 1;            // Gram row tile (source classes)
  int tj = wave & 1;             // Gram col tile (convert classes)

  const float* CS = ws + CEN_OFF;                 // source centers [19][256]
  const float* CC = ws + CEN_OFF + NCLASS * CHN;  // convert centers [19][256]

  int mrow = ti * 16 + (lane & 15);
  int nrow = tj * 16 + (lane & 15);
  float amask = (mrow < NCLASS) ? 1.0f : 0.0f;    // branchless zero padding
  float bmask = (nrow < NCLASS) ? 1.0f : 0.0f;
  int mi = (mrow < NCLASS) ? mrow : 0;
  int ni = (nrow < NCLASS) ? nrow : 0;
  int khalf = (lane >> 4) * 2;                    // lanes 0-15: K=0,1; 16-31: K=2,3

  v8f c = {0.f, 0.f, 0.f, 0.f, 0.f, 0.f, 0.f, 0.f};
  for (int k = 0; k < CHN; k += 4) {
    v2f a, b;
    a.x = CS[mi * CHN + k + khalf]     * amask;
    a.y = CS[mi * CHN + k + khalf + 1] * amask;
    b.x = CC[ni * CHN + k + khalf]     * bmask;   // B[p][n] = CC[n][k+p]
    b.y = CC[ni * CHN + k + khalf + 1] * bmask;
    c = __builtin_amdgcn_wmma_f32_16x16x4_f32(false, a, false, b,
                                              (short)0, c, false, false);
  }

  // C/D layout: VGPR v -> M = v (lanes 0-15) / v+8 (lanes 16-31), N = lane%16.
#pragma unroll
  for (int v = 0; v < 8; ++v) {
    int M = (lane < 16) ? v : v + 8;
    G[(ti * 16 + M) * 33 + tj * 16 + (lane & 15)] = c[v];
  }
  __syncthreads();

  const float* ssq = ws + SS_OFF;   // [0..18]=source, [19..37]=convert
  const float* s1  = ws + S1_OFF;
  const float cterm = (float)CHN * EPS_T * EPS_T;

  float local = 0.0f;
  for (int q = threadIdx.x; q < NCLASS * (NCLASS - 1); q += blockDim.x) {
    int i  = q / (NCLASS - 1);
    int jj = q % (NCLASS - 1);
    int j  = jj + (jj >= i ? 1 : 0);
    // ||a - b + eps*1||^2 = Sa2 + Sb2 - 2 a.b + 2 eps (sum a - sum b) + C eps^2
    float dap2 = ssq[i] + ssq[NCLASS + i] - 2.0f * G[i * 33 + i]
               + 2.0f * EPS_T * (s1[i] - s1[NCLASS + i]) + cterm;
    float dan2 = ssq[i] + ssq[NCLASS + j] - 2.0f * G[i * 33 + j]
               + 2.0f * EPS_T * (s1[i] - s1[NCLASS + j]) + cterm;
    float dap = sqrtf(fmaxf(dap2, 0.0f));
    float dan = sqrtf(fmaxf(dan2, 0.0f));
    local += fmaxf(dap - dan + MARGIN, 0.0f);
  }
  part[threadIdx.x] = local;
  __syncthreads();
  for (int s = 64; s > 0; s >>= 1) {
    if ((int)threadIdx.x < s) part[threadIdx.x] += part[threadIdx.x + s];
    __syncthreads();
  }
  if (threadIdx.x == 0) out[0] = part[0] / (float)(NCLASS * (NCLASS - 1));
}

extern "C" void kernel_launch(void* const* d_in, const int* in_sizes, int n_in,
                              void* d_out, int out_size, void* d_ws, size_t ws_size,
                              hipStream_t stream) {
  (void)in_sizes; (void)n_in; (void)out_size; (void)ws_size;
  const float* fs  = (const float*)d_in[0];
  const float* fc  = (const float*)d_in[1];
  const int*   seg = (const int*)d_in[2];
  float* ws  = (float*)d_ws;
  float* out = (float*)d_out;

  zero_acc   <<<(NROWS * CHN + 255) / 256, 256, 0, stream>>>(ws);
  norm_scatter<<<NPIX / 256, 256, 0, stream>>>(fs, fc, seg, ws);
  centers    <<<1, 256, 0, stream>>>(ws);
  gram_loss  <<<1, 128, 0, stream>>>(ws, out);
}